// GCN_79577154060348
// MI455X (gfx1250) — compile-verified
//
#include <hip/hip_runtime.h>
#include <hip/hip_bf16.h>

typedef __attribute__((ext_vector_type(2))) float v2f;
typedef __attribute__((ext_vector_type(8))) float v8f;

#define D_IN  128
#define D_HID 128
#define D_OUTD 64
#define BN_EPS 1e-5f

__device__ __forceinline__ void atomic_add_f32(float* p, float v) {
    // hardware global_atomic_add_f32 (fire-and-forget, no return)
    unsafeAtomicAdd(p, v);
}

// ---------------- init: deg = 1 (self loop), h1 = 0, out = 0 ----------------
__global__ void k_init(float* __restrict__ deg, float* __restrict__ h1,
                       float* __restrict__ out, int NN, long h1n, long outn) {
    long i = (long)blockIdx.x * blockDim.x + threadIdx.x;
    if (i < NN)   deg[i] = 1.0f;
    if (i < h1n)  h1[i]  = 0.0f;
    if (i < outn) out[i] = 0.0f;
}

// ---------------- deg accumulation over edges ----------------
__global__ void k_deg(const int* __restrict__ dst, const float* __restrict__ ew,
                      float* __restrict__ deg, int E) {
    int e = blockIdx.x * blockDim.x + threadIdx.x;
    if (e >= E) return;
    atomic_add_f32(&deg[dst[e]], ew[e]);
}

// ---------------- dinv = rsqrt(deg) ----------------
__global__ void k_dinv(const float* __restrict__ deg, float* __restrict__ dinv, int NN) {
    int i = blockIdx.x * blockDim.x + threadIdx.x;
    if (i >= NN) return;
    float d = deg[i];
    dinv[i] = d > 0.0f ? __frsqrt_rn(d) : 0.0f;
}

// ---------------- fp32 WMMA GEMM: C[M,N] = A[M,K] @ B[K,N] ----------------
// Block = N/16 waves; wave w computes the 16x16 tile (blockIdx.x, w).
// B is staged block-wide into LDS with async copies (ASYNCcnt), then the
// K-loop runs V_WMMA_F32_16X16X4_F32 fed by immediate-offset global (A) and
// LDS (B) loads.
// A 16x4 layout : VGPR v, lane = half*16 + m  -> A[m][v + 2*half]
// B 4x16 layout : VGPR v, lane = half*16 + n  -> B[v + 2*half][n]
// C/D 16x16     : VGPR r, lane = half*16 + n  -> C[r + 8*half][n]
template<int K, int N>
__global__ void k_gemm_wmma_f32(const float* __restrict__ A, const float* __restrict__ B,
                                float* __restrict__ C, int M) {
    __shared__ float Bs[K * N];                 // single shared array -> LDS offset 0
    const int tid = threadIdx.x;
    const int nthreads = (N / 16) * 32;         // == blockDim.x

    // ---- async-stage B (K*N floats) into LDS, 16B per lane per issue ----
    constexpr int CHUNKS = K * N / 4;           // number of 16-byte chunks
    for (int c = tid; c < CHUNKS; c += nthreads) {
        unsigned lds_off = (unsigned)c * 16u;   // Bs starts at LDS offset 0
        const float* g = B + (size_t)c * 4;
        asm volatile("global_load_async_to_lds_b128 %0, %1, off"
                     :: "v"(lds_off), "v"(g) : "memory");
    }
    asm volatile("s_wait_asynccnt 0x0" ::: "memory");
    __syncthreads();

    const int wave = tid >> 5;                  // wave32
    const int lane = tid & 31;
    const int half = lane >> 4;
    const int mr   = lane & 15;

    int row = blockIdx.x * 16 + mr;
    if (row >= M) row = M - 1;                  // clamp: EXEC stays all-ones
    const int col = wave * 16 + mr;

    const float* ap = A + (size_t)row * K + 2 * half;   // A[row][2*half + k]
    const float* bp = Bs + (2 * half) * N + col;        // Bs[2*half + k][col]

    v8f acc = {};
    #pragma unroll
    for (int k = 0; k < K; k += 4) {
        v2f a = *(const v2f*)(ap + k);          // global, imm offset
        v2f b;
        b.x = bp[k * N];                        // ds_load, imm offset
        b.y = bp[k * N + N];
        acc = __builtin_amdgcn_wmma_f32_16x16x4_f32(
            /*neg_a=*/false, a, /*neg_b=*/false, b,
            /*c_mod=*/(short)0, acc, /*reuse_a=*/false, /*reuse_b=*/false);
    }
    #pragma unroll
    for (int r = 0; r < 8; ++r) {
        int m = blockIdx.x * 16 + r + 8 * half;
        if (m < M) C[(size_t)m * N + col] = acc[r];
    }
}

// ---------------- edge scatter: one wave per edge ----------------
// out[dst] += xw[src] * (dinv[src]*ew*dinv[dst])
template<int D>
__global__ void k_scatter(const float* __restrict__ xw, const int* __restrict__ src,
                          const int* __restrict__ dst, const float* __restrict__ ew,
                          const float* __restrict__ dinv, float* __restrict__ out,
                          int E) {
    int wave = (blockIdx.x * blockDim.x + threadIdx.x) >> 5;
    int lane = threadIdx.x & 31;
    if (wave >= E) return;
    const int s = src[wave];
    const int d = dst[wave];
    const float* xs = xw + (size_t)s * D;
    __builtin_prefetch(xs, 0, 1);               // global_prefetch_b8
    const float norm = dinv[s] * ew[wave] * dinv[d];
    float* od = out + (size_t)d * D;
    if (D == 128) {                             // one b128 load + 4 atomics per lane
        float4 v = *(const float4*)(xs + lane * 4);
        float* p = od + lane * 4;
        atomic_add_f32(p + 0, v.x * norm);
        atomic_add_f32(p + 1, v.y * norm);
        atomic_add_f32(p + 2, v.z * norm);
        atomic_add_f32(p + 3, v.w * norm);
    } else {                                    // D == 64: one b64 load + 2 atomics
        float2 v = *(const float2*)(xs + lane * 2);
        float* p = od + lane * 2;
        atomic_add_f32(p + 0, v.x * norm);
        atomic_add_f32(p + 1, v.y * norm);
    }
}

// ---------------- self-loop + bias + BN(eval) + ReLU (in place on h) ----------------
__global__ void k_selfloop_bn_relu(float* __restrict__ h, const float* __restrict__ xw,
                                   const float* __restrict__ dinv,
                                   const float* __restrict__ b1,
                                   const float* __restrict__ gamma,
                                   const float* __restrict__ beta,
                                   const float* __restrict__ mean,
                                   const float* __restrict__ var,
                                   int NN, int D) {
    long i = (long)blockIdx.x * blockDim.x + threadIdx.x;
    if (i >= (long)NN * D) return;
    int node = (int)(i / D);
    int c    = (int)(i % D);
    float di = dinv[node];
    float v = h[i] + xw[i] * di * di + b1[c];                 // self-loop norm = dinv^2
    v = (v - mean[c]) * __frsqrt_rn(var[c] + BN_EPS) * gamma[c] + beta[c];
    h[i] = v > 0.0f ? v : 0.0f;
}

// ---------------- self-loop + bias for layer 2 output ----------------
__global__ void k_selfloop_bias(float* __restrict__ out, const float* __restrict__ hw,
                                const float* __restrict__ dinv,
                                const float* __restrict__ b2, int NN, int D) {
    long i = (long)blockIdx.x * blockDim.x + threadIdx.x;
    if (i >= (long)NN * D) return;
    int node = (int)(i / D);
    int c    = (int)(i % D);
    float di = dinv[node];
    out[i] += hw[i] * di * di + b2[c];
}

extern "C" void kernel_launch(void* const* d_in, const int* in_sizes, int n_in,
                              void* d_out, int out_size, void* d_ws, size_t ws_size,
                              hipStream_t stream) {
    const float* x     = (const float*)d_in[0];
    const int*   ei    = (const int*)  d_in[1];
    const float* ew    = (const float*)d_in[2];
    const float* W1    = (const float*)d_in[3];
    const float* b1    = (const float*)d_in[4];
    const float* gamma = (const float*)d_in[5];
    const float* beta  = (const float*)d_in[6];
    const float* rmean = (const float*)d_in[7];
    const float* rvar  = (const float*)d_in[8];
    const float* W2    = (const float*)d_in[9];
    const float* b2    = (const float*)d_in[10];
    float* out = (float*)d_out;

    const int NN = in_sizes[0] / D_IN;   // 50000
    const int E  = in_sizes[2];          // 800000
    const int* src = ei;
    const int* dst = ei + E;

    // workspace layout (floats): xw[NN*128] | h1[NN*128] | hw[NN*64] | deg[NN] | dinv[NN]
    float* ws   = (float*)d_ws;
    float* xw   = ws;
    float* h1   = xw + (size_t)NN * D_HID;
    float* hw   = h1 + (size_t)NN * D_HID;
    float* deg  = hw + (size_t)NN * D_OUTD;
    float* dinv = deg + NN;

    const long h1n  = (long)NN * D_HID;
    const long outn = (long)NN * D_OUTD;

    // 1) init
    k_init<<<dim3((unsigned)((h1n + 255) / 256)), dim3(256), 0, stream>>>(deg, h1, out, NN, h1n, outn);
    // 2) degree accumulation
    k_deg<<<dim3((E + 255) / 256), dim3(256), 0, stream>>>(dst, ew, deg, E);
    // 3) dinv
    k_dinv<<<dim3((NN + 255) / 256), dim3(256), 0, stream>>>(deg, dinv, NN);
    // 4) GEMM1: xw = x @ W1   (M=NN, K=128, N=128) — 8 waves/block
    k_gemm_wmma_f32<D_IN, D_HID><<<dim3((NN + 15) / 16), dim3(256), 0, stream>>>(x, W1, xw, NN);
    // 5) edge scatter layer 1: one wave per edge
    k_scatter<D_HID><<<dim3((E + 7) / 8), dim3(256), 0, stream>>>(xw, src, dst, ew, dinv, h1, E);
    // 6) self-loop + b1 + BN + ReLU (in place)
    k_selfloop_bn_relu<<<dim3((unsigned)((h1n + 255) / 256)), dim3(256), 0, stream>>>(
        h1, xw, dinv, b1, gamma, beta, rmean, rvar, NN, D_HID);
    // 7) GEMM2: hw = h1 @ W2  (M=NN, K=128, N=64) — 4 waves/block
    k_gemm_wmma_f32<D_HID, D_OUTD><<<dim3((NN + 15) / 16), dim3(128), 0, stream>>>(h1, W2, hw, NN);
    // 8) edge scatter layer 2 into out
    k_scatter<D_OUTD><<<dim3((E + 7) / 8), dim3(256), 0, stream>>>(hw, src, dst, ew, dinv, out, E);
    // 9) self-loop + b2
    k_selfloop_bias<<<dim3((unsigned)((outn + 255) / 256)), dim3(256), 0, stream>>>(out, hw, dinv, b2, NN, D_OUTD);
}